// GINModel_2645699854678
// MI455X (gfx1250) — compile-verified
//
#include <hip/hip_runtime.h>
#include <hip/hip_bf16.h>

#define N_NODES 100000
#define N_EDGES 1600000
#define HIDDEN  128
#define OUTDIM  128
#define N_LAYERS 2
#define N_GRAPHS 64
#define SCAN_BLOCKS ((N_NODES + 255) / 256)   // 391

typedef float v2f __attribute__((ext_vector_type(2)));
typedef float v8f __attribute__((ext_vector_type(8)));

// ---------------------------------------------------------------- utilities

__global__ void k_zero4(float4* p, int n4) {
    int t = blockIdx.x * blockDim.x + threadIdx.x;
    int stride = gridDim.x * blockDim.x;
    float4 z = make_float4(0.f, 0.f, 0.f, 0.f);
    for (int i = t; i < n4; i += stride) p[i] = z;
}

// h0[node] = emb[nfeats[node]]  (one wave per node, float4 per lane)
__global__ void k_embed(const int* __restrict__ nf,
                        const float* __restrict__ emb,
                        float* __restrict__ h) {
    int t = blockIdx.x * blockDim.x + threadIdx.x;
    int node = t >> 5;
    int lane = t & 31;
    if (node >= N_NODES) return;
    const float4* s = (const float4*)(emb + (size_t)nf[node] * HIDDEN);
    float4* d = (float4*)(h + (size_t)node * HIDDEN);
    d[lane] = s[lane];
}

// ------------------------------------------------------------- CSR-by-dst
// Rebuilt every call (stateless/deterministic work); replaces 410M f32
// atomics with ~3.2M int atomics + streaming gather.

__global__ void k_degree(const int* __restrict__ dst, int* __restrict__ deg) {
    int e = blockIdx.x * blockDim.x + threadIdx.x;
    if (e < N_EDGES) atomicAdd(&deg[dst[e]], 1);
}

__global__ void k_block_sums(const int* __restrict__ deg,
                             int* __restrict__ partial) {
    __shared__ int s[256];
    int t = threadIdx.x;
    int idx = blockIdx.x * 256 + t;
    s[t] = (idx < N_NODES) ? deg[idx] : 0;
    __syncthreads();
    for (int off = 128; off > 0; off >>= 1) {
        if (t < off) s[t] += s[t + off];
        __syncthreads();
    }
    if (t == 0) partial[blockIdx.x] = s[0];
}

// sequential exclusive scan of the 391 block partials (trivial work)
__global__ void k_scan_partials(int* __restrict__ partial, int nb,
                                int* __restrict__ ptr_last) {
    if (blockIdx.x == 0 && threadIdx.x == 0) {
        int run = 0;
        for (int i = 0; i < nb; ++i) {
            int v = partial[i];
            partial[i] = run;
            run += v;
        }
        *ptr_last = run;   // ptr[N_NODES] == N_EDGES
    }
}

// per-block Hillis-Steele exclusive scan -> ptr[] and cursor[]
__global__ void k_block_scan(const int* __restrict__ deg,
                             const int* __restrict__ partial,
                             int* __restrict__ ptr,
                             int* __restrict__ cursor) {
    __shared__ int s[256];
    int t = threadIdx.x;
    int idx = blockIdx.x * 256 + t;
    int v = (idx < N_NODES) ? deg[idx] : 0;
    s[t] = v;
    __syncthreads();
    for (int off = 1; off < 256; off <<= 1) {
        int u = (t >= off) ? s[t - off] : 0;
        __syncthreads();
        s[t] += u;
        __syncthreads();
    }
    if (idx < N_NODES) {
        int e = partial[blockIdx.x] + s[t] - v;   // exclusive prefix
        ptr[idx]    = e;
        cursor[idx] = e;
    }
}

__global__ void k_fill(const int* __restrict__ src,
                       const int* __restrict__ dst,
                       int* __restrict__ cursor,
                       int* __restrict__ ecsr) {
    int e = blockIdx.x * blockDim.x + threadIdx.x;
    if (e >= N_EDGES) return;
    int slot = atomicAdd(&cursor[dst[e]], 1);
    ecsr[slot] = src[e];
}

// x[n] = h[n] + sum_{s in N_in(n)} h[s]   (GIN eps=0 self-term fused)
// one wave per node; each neighbor row is a coalesced 512B load.
__global__ void k_aggregate(const float* __restrict__ h,
                            const int* __restrict__ ptr,
                            const int* __restrict__ ecsr,
                            float* __restrict__ x) {
    int t = blockIdx.x * blockDim.x + threadIdx.x;
    int node = t >> 5;
    int lane = t & 31;
    if (node >= N_NODES) return;
    int beg = ptr[node];
    int end = ptr[node + 1];
    float4 acc = ((const float4*)(h + (size_t)node * HIDDEN))[lane];
    for (int i = beg; i < end; ++i) {
        int s = ecsr[i];
        if (i + 1 < end)   // cover latency of the next random row
            __builtin_prefetch(h + (size_t)ecsr[i + 1] * HIDDEN, 0, 0);
        float4 v = ((const float4*)(h + (size_t)s * HIDDEN))[lane];
        acc.x += v.x; acc.y += v.y; acc.z += v.z; acc.w += v.w;
    }
    ((float4*)(x + (size_t)node * HIDDEN))[lane] = acc;
}

// ------------------------------------------------ WMMA f32 16x16x4 GEMM core
// hout = leaky_relu( x @ W + bias ),  M = N_NODES, K = N = 128.
__global__ void k_combine(const float* __restrict__ x,
                          const float* __restrict__ W,
                          const float* __restrict__ bias,
                          float* __restrict__ hout) {
    __shared__ float sA[16][HIDDEN + 4];
    int tid  = threadIdx.x;
    int wave = tid >> 5;
    int lane = tid & 31;
    int half = lane >> 4;       // 0: K pair {0,1}, 1: K pair {2,3}
    int l16  = lane & 15;
    int nb   = wave * 16;       // N-tile base

    v2f breg[32];
#pragma unroll
    for (int ks = 0; ks < 32; ++ks) {
        int k = ks * 4 + 2 * half;
        breg[ks].x = W[(k + 0) * HIDDEN + nb + l16];
        breg[ks].y = W[(k + 1) * HIDDEN + nb + l16];
    }
    float bv = bias[nb + l16];

    for (int mt = blockIdx.x; mt < N_NODES / 16; mt += gridDim.x) {
        size_t base = (size_t)mt * 16 * HIDDEN;
        for (int i = tid; i < 16 * (HIDDEN / 4); i += blockDim.x) {
            int r  = i >> 5;
            int c4 = i & 31;
            ((float4*)&sA[r][c4 * 4])[0] =
                ((const float4*)(x + base + r * HIDDEN))[c4];
        }
        __syncthreads();

        v8f c;
#pragma unroll
        for (int j = 0; j < 8; ++j) c[j] = bv;

#pragma unroll
        for (int ks = 0; ks < 32; ++ks) {
            int k = ks * 4 + 2 * half;
            v2f a;
            a.x = sA[l16][k + 0];
            a.y = sA[l16][k + 1];
            c = __builtin_amdgcn_wmma_f32_16x16x4_f32(
                    false, a, false, breg[ks], (short)0, c, false, false);
        }

#pragma unroll
        for (int j = 0; j < 8; ++j) {
            int row = mt * 16 + j + half * 8;
            float v = c[j];
            v = (v >= 0.f) ? v : 0.2f * v;
            hout[(size_t)row * HIDDEN + nb + l16] = v;
        }
        __syncthreads();
    }
}

// JK readout: per layer l (blockIdx.y), out[g] += h_l @ read_W[l] + read_b[l]
__global__ void k_readout(const float* __restrict__ h0,
                          const float* __restrict__ h1,
                          const float* __restrict__ h2,
                          const float* __restrict__ RW,
                          const float* __restrict__ Rb,
                          const int* __restrict__ gids,
                          float* __restrict__ out) {
    __shared__ float sA[16][HIDDEN + 4];
    int layer = blockIdx.y;
    const float* h    = (layer == 0) ? h0 : (layer == 1) ? h1 : h2;
    const float* W    = RW + (size_t)layer * HIDDEN * OUTDIM;
    const float* bias = Rb + (size_t)layer * OUTDIM;

    int tid  = threadIdx.x;
    int wave = tid >> 5;
    int lane = tid & 31;
    int half = lane >> 4;
    int l16  = lane & 15;
    int nb   = wave * 16;

    v2f breg[32];
#pragma unroll
    for (int ks = 0; ks < 32; ++ks) {
        int k = ks * 4 + 2 * half;
        breg[ks].x = W[(k + 0) * OUTDIM + nb + l16];
        breg[ks].y = W[(k + 1) * OUTDIM + nb + l16];
    }
    float bv = bias[nb + l16];

    for (int mt = blockIdx.x; mt < N_NODES / 16; mt += gridDim.x) {
        size_t base = (size_t)mt * 16 * HIDDEN;
        for (int i = tid; i < 16 * (HIDDEN / 4); i += blockDim.x) {
            int r  = i >> 5;
            int c4 = i & 31;
            ((float4*)&sA[r][c4 * 4])[0] =
                ((const float4*)(h + base + r * HIDDEN))[c4];
        }
        __syncthreads();

        v8f c;
#pragma unroll
        for (int j = 0; j < 8; ++j) c[j] = bv;

#pragma unroll
        for (int ks = 0; ks < 32; ++ks) {
            int k = ks * 4 + 2 * half;
            v2f a;
            a.x = sA[l16][k + 0];
            a.y = sA[l16][k + 1];
            c = __builtin_amdgcn_wmma_f32_16x16x4_f32(
                    false, a, false, breg[ks], (short)0, c, false, false);
        }

        // fused sum pooling: only 38.4M f32 atomics total (12.8M per layer), into 8K slots
#pragma unroll
        for (int j = 0; j < 8; ++j) {
            int row = mt * 16 + j + half * 8;
            int g = gids[row];
            unsafeAtomicAdd(out + (size_t)g * OUTDIM + nb + l16, c[j]);
        }
        __syncthreads();
    }
}

// ------------------------------------------------------------------ launcher

extern "C" void kernel_launch(void* const* d_in, const int* in_sizes, int n_in,
                              void* d_out, int out_size, void* d_ws, size_t ws_size,
                              hipStream_t stream) {
    const int*   nfeats = (const int*)  d_in[0];
    // d_in[1] = efeats (unused by the reference)
    const int*   src    = (const int*)  d_in[2];
    const int*   dst    = (const int*)  d_in[3];
    const int*   gids   = (const int*)  d_in[4];
    // d_in[5] = num_graphs (scalar, fixed 64)
    const float* emb    = (const float*)d_in[6];
    const float* comb_W = (const float*)d_in[7];
    const float* comb_b = (const float*)d_in[8];
    const float* read_W = (const float*)d_in[9];
    const float* read_b = (const float*)d_in[10];
    float*       out    = (float*)d_out;

    const size_t feat_elems = (size_t)N_NODES * HIDDEN;      // 12.8M floats
    float* h0 = (float*)d_ws;
    float* h1 = h0 + feat_elems;
    float* h2 = h1 + feat_elems;
    float* xb = h2 + feat_elems;                 // aggregated input buffer
    int*   ecsr    = (int*)(xb + feat_elems);    // 1.6M
    int*   ptr     = ecsr + N_EDGES;             // 100001
    int*   cursor  = ptr + (N_NODES + 1);        // 100000
    int*   deg     = cursor + N_NODES;           // 100000
    int*   partial = deg + N_NODES;              // 391

    const int ZB = 256;
    const int EB = (N_EDGES + ZB - 1) / ZB;      // 6250

    // zero output accumulator + degree histogram
    k_zero4<<<32, ZB, 0, stream>>>((float4*)out, out_size / 4);
    k_zero4<<<128, ZB, 0, stream>>>((float4*)deg, N_NODES / 4);

    // node encoding
    k_embed<<<(N_NODES * 32) / ZB, ZB, 0, stream>>>(nfeats, emb, h0);

    // CSR-by-dst build (reused by both layers)
    k_degree<<<EB, ZB, 0, stream>>>(dst, deg);
    k_block_sums<<<SCAN_BLOCKS, ZB, 0, stream>>>(deg, partial);
    k_scan_partials<<<1, 32, 0, stream>>>(partial, SCAN_BLOCKS, ptr + N_NODES);
    k_block_scan<<<SCAN_BLOCKS, ZB, 0, stream>>>(deg, partial, ptr, cursor);
    k_fill<<<EB, ZB, 0, stream>>>(src, dst, cursor, ecsr);

    // layer 0
    k_aggregate<<<(N_NODES * 32) / ZB, ZB, 0, stream>>>(h0, ptr, ecsr, xb);
    k_combine<<<1024, ZB, 0, stream>>>(xb, comb_W, comb_b, h1);

    // layer 1
    k_aggregate<<<(N_NODES * 32) / ZB, ZB, 0, stream>>>(h1, ptr, ecsr, xb);
    k_combine<<<1024, ZB, 0, stream>>>(xb, comb_W + HIDDEN * HIDDEN,
                                       comb_b + HIDDEN, h2);

    // JK readout + pooling (blockIdx.y = layer 0..2)
    dim3 rg(1024, N_LAYERS + 1);
    k_readout<<<rg, ZB, 0, stream>>>(h0, h1, h2, read_W, read_b, gids, out);
}